// MicroSlot_52974126629395
// MI455X (gfx1250) — compile-verified
//
#include <hip/hip_runtime.h>
#include <cstdint>

// MicroSlot fused slot-attention video model for MI455X (gfx1250, wave32).
// One wave per batch element; all 16x16 GEMMs via V_WMMA_F32_16X16X4_F32 chains.
// Frames are DMA'd into LDS with the Tensor Data Mover (double-buffered,
// TENSORcnt-gated); everything else stays resident in LDS/registers.

typedef __attribute__((ext_vector_type(2))) float v2f;
typedef __attribute__((ext_vector_type(8))) float v8f;
typedef __attribute__((ext_vector_type(4))) unsigned int u32x4;
typedef __attribute__((ext_vector_type(8))) int i32x8;
typedef __attribute__((ext_vector_type(4))) int i32x4;

#define DEV __device__ __forceinline__

DEV v8f splat8(float x) {
  v8f r;
#pragma unroll
  for (int i = 0; i < 8; ++i) r[i] = x;
  return r;
}

// ---- WMMA fragment loaders (wave32 layouts per CDNA5 ISA 7.12.2) ----
// A 16x4 f32: lanes0-15 m=lane, vgpr0=k, vgpr1=k+1; lanes16-31 m=lane-16, k+2/k+3.
// B 4x16 f32: mirrored (n across lanes, k across vgprs/half).
// C/D 16x16 f32: lanes0-15 rows v, lanes16-31 rows v+8, col = lane&15.

// fragment of B from B[k][n] row-major stride16 (also serves as A-fragment of T^T)
DEV v2f ldKxN(const float* T, int kc) {
  const int n = threadIdx.x & 15;
  const int k0 = kc + ((threadIdx.x & 16) >> 3);
  return v2f{T[k0 * 16 + n], T[(k0 + 1) * 16 + n]};
}
// fragment of A from A[m][k] row-major stride16 (also serves as B-fragment of T^T)
DEV v2f ldMxK(const float* T, int kc) {
  const int m = threadIdx.x & 15;
  const int k0 = kc + ((threadIdx.x & 16) >> 3);
  return v2f{T[m * 16 + k0], T[m * 16 + k0 + 1]};
}
DEV v2f ldMxK64(const float* T, int kc) {  // A[m][k], stride 64 (mix_w 16x64)
  const int m = threadIdx.x & 15;
  const int k0 = kc + ((threadIdx.x & 16) >> 3);
  return v2f{T[m * 64 + k0], T[m * 64 + k0 + 1]};
}
DEV v2f ldPad4(const float* T, int kc) {  // A with rows 4..15 == 0 (slot tiles)
  const int m = threadIdx.x & 15;
  const int k0 = kc + ((threadIdx.x & 16) >> 3);
  if (m < 4) return v2f{T[m * 16 + k0], T[m * 16 + k0 + 1]};
  return v2f{0.f, 0.f};
}
// s2s B fragment straight from global: B[d][n] = s2s_w[(q*16+n)*16+d]
DEV v2f ldS2s(const float* __restrict__ G, int q, int kc) {
  const int n = threadIdx.x & 15;
  const int k0 = kc + ((threadIdx.x & 16) >> 3);
  const float* p = G + (q * 16 + n) * 16 + k0;
  return v2f{p[0], p[1]};
}
DEV float frpix(const float* F, int m, int k) {  // patch m=(i*4+j), pixel k=(di*4+dj)
  return F[(4 * (m >> 2) + (k >> 2)) * 16 + 4 * (m & 3) + (k & 3)];
}
DEV v2f ldFrame(const float* F, int kc) {
  const int m = threadIdx.x & 15;
  const int k0 = kc + ((threadIdx.x & 16) >> 3);
  return v2f{frpix(F, m, k0), frpix(F, m, k0 + 1)};
}
DEV void stC(float* T, const v8f& c) {  // full C tile -> row-major [16][16]
  const int n = threadIdx.x & 15;
  const int mo = (threadIdx.x & 16) >> 1;
#pragma unroll
  for (int v = 0; v < 8; ++v) T[(v + mo) * 16 + n] = c[v];
}
DEV void stC4(float* T, const v8f& c) {  // rows 0..3 only (slot-space results)
  if (threadIdx.x & 16) return;
  const int n = threadIdx.x & 15;
#pragma unroll
  for (int v = 0; v < 4; ++v) T[v * 16 + n] = c[v];
}

// ---- Tensor Data Mover: async 1KB tile global -> LDS (D# per ISA ch.8) ----
// 6-arg builtin variant: (g0, g1, g2, g3, g_extra, cpol)
DEV void tdm_load_1kb(unsigned lds_byte_off, const float* __restrict__ gptr) {
  const uint64_t ga = (uint64_t)(uintptr_t)gptr;
  u32x4 g0;
  g0[0] = 1u;                                   // count=1, user mode
  g0[1] = lds_byte_off;                         // lds_addr
  g0[2] = (unsigned)(ga & 0xffffffffu);         // global_addr[31:0]
  g0[3] = (unsigned)((ga >> 32) & 0x01ffffffu)  // global_addr[56:32]
          | (2u << 30);                         // type=2 (image)
  i32x8 g1;
  g1[0] = (int)(3u << 16);     // data_size=8B, workgroup_mask=0, no flags
  g1[1] = (int)(128u << 16);   // tensor_dim0[15:0]=128 (bits 63:48)
  g1[2] = 0;                   // tensor_dim0 hi / tensor_dim1 lo
  g1[3] = (int)(128u << 16);   // tile_dim0=128 (bits 127:112) -> 1KB
  g1[4] = 0;                   // tile_dim1=0, tile_dim2=0 (1-D tile)
  g1[5] = 128;                 // tensor_dim0_stride
  g1[6] = 0;
  g1[7] = 0;
  const i32x4 z4 = {0, 0, 0, 0};  // groups 2/3 unused (<=2D)
  const i32x8 z8 = {0, 0, 0, 0, 0, 0, 0, 0};
  __builtin_amdgcn_tensor_load_to_lds(g0, g1, z4, z4, z8, 0);
}
DEV unsigned lds_off(const void* p) {  // flat LDS addr[31:0] == byte offset
  return (unsigned)(uintptr_t)p;
}

#define WMMA4(acc, a, b)                                                      \
  acc = __builtin_amdgcn_wmma_f32_16x16x4_f32(false, (a), false, (b),         \
                                              (short)0, (acc), false, false)

#define MM16(acc, AEXPR, BEXPR)          \
  do {                                   \
    _Pragma("unroll")                    \
    for (int kc = 0; kc < 16; kc += 4) { \
      v2f a_ = (AEXPR);                  \
      v2f b_ = (BEXPR);                  \
      WMMA4(acc, a_, b_);                \
    }                                    \
  } while (0)

#define MM64(acc, AEXPR, BEXPR)          \
  do {                                   \
    _Pragma("unroll")                    \
    for (int kc = 0; kc < 64; kc += 4) { \
      v2f a_ = (AEXPR);                  \
      v2f b_ = (BEXPR);                  \
      WMMA4(acc, a_, b_);                \
    }                                    \
  } while (0)

DEV float sigm(float x) { return 1.f / (1.f + __expf(-x)); }

__global__ __launch_bounds__(128, 4) void microslot_kernel(
    const float* __restrict__ frames, const float* __restrict__ enc_w,
    const float* __restrict__ enc_b, const float* __restrict__ ln_w,
    const float* __restrict__ ln_b, const float* __restrict__ slot_mu,
    const float* __restrict__ wq, const float* __restrict__ wk,
    const float* __restrict__ wv, const float* __restrict__ wih,
    const float* __restrict__ whh, const float* __restrict__ bih,
    const float* __restrict__ bhh, const float* __restrict__ s2s_w,
    const float* __restrict__ s2s_b, const float* __restrict__ mix_w,
    const float* __restrict__ mix_b, const float* __restrict__ dec_w,
    const float* __restrict__ dec_b, float* __restrict__ out) {
  constexpr int T = 8, TS = 6;
  // shared weights, pre-transposed into B[k][n] layout (s2s stays in global/L2)
  __shared__ __align__(16) float sEnc[256], sWk[256], sWv[256], sWq[256];
  __shared__ __align__(16) float sWih[768], sWhh[768], sMix[1024], sDec[256];
  // per-wave scratch + double-buffered TDM frame tiles
  __shared__ __align__(16) float sKK[4][256], sVV[4][256], sSL[4][64];
  __shared__ __align__(16) float sSCR[4][256], sMXD[4][1024];
  __shared__ __align__(16) float sFRM[4][2][256];

  const int tid = threadIdx.x;
  for (int i = tid; i < 256; i += 128) {
    const int r = i >> 4, c = i & 15;
    sEnc[i] = enc_w[c * 16 + r];  // B[pixel][cout] = enc_w[cout][pixel]
    sWk[i] = wk[c * 16 + r];      // B[cin][d] = wk[d][cin]
    sWv[i] = wv[c * 16 + r];
    sWq[i] = wq[c * 16 + r];
    sDec[i] = dec_w[i];           // B[c][de] = dec_w[c][de]
  }
  for (int i = tid; i < 768; i += 128) {
    const int g = i >> 8, r = (i >> 4) & 15, c = i & 15;
    sWih[i] = wih[(g * 16 + c) * 16 + r];  // B[d][gate-col] per gate chunk
    sWhh[i] = whh[(g * 16 + c) * 16 + r];
  }
  for (int i = tid; i < 1024; i += 128) sMix[i] = mix_w[i];  // A: 16x64 rm
  __syncthreads();

  const int wid = tid >> 5, lane = tid & 31, l16 = lane & 15;
  const int mo = (lane & 16) >> 1;  // +8 rows for hi half of wave
  const size_t b = (size_t)blockIdx.x * 4 + wid;

  float* KK = sKK[wid];
  float* VV = sVV[wid];
  float* SL = sSL[wid];
  float* SCR = sSCR[wid];
  float* MXD = sMXD[wid];
  float* FRM0 = sFRM[wid][0];
  float* FRM1 = sFRM[wid][1];

  const float lnw = ln_w[l16], lnb = ln_b[l16];
  const float encb = enc_b[l16];
  const float bihr = bih[l16], bihz = bih[16 + l16], bihn = bih[32 + l16];
  const float bhhr = bhh[l16], bhhz = bhh[16 + l16], bhhn = bhh[32 + l16];
  const float decb = dec_b[0];
  float mixb0[8];
#pragma unroll
  for (int v = 0; v < 8; ++v) mixb0[v] = mix_b[v + mo];

  for (int i = lane; i < 64; i += 32) SL[i] = slot_mu[i];  // broadcast slots

  const float* fbase = frames + b * T * 256;
  // TDM pipeline prologue: frame index 1 -> buffer 0
  tdm_load_1kb(lds_off(FRM0), fbase + 1 * 256);

  for (int t = 0; t < TS; ++t) {
    // issue next frame DMA, then wait for the current one (in-order per wave)
    if (t + 1 < TS) {
      tdm_load_1kb(lds_off(((t + 1) & 1) ? FRM1 : FRM0),
                   fbase + (size_t)(t + 2) * 256);
      __builtin_amdgcn_s_wait_tensorcnt((short)1);
    } else {
      __builtin_amdgcn_s_wait_tensorcnt((short)0);
    }
    const float* F = (t & 1) ? FRM1 : FRM0;

    // encoder conv(k4,s4): feat[n][c] = GELU(patch[n][p] @ enc_w^T + b)
    v8f x = splat8(encb);
    MM16(x, ldFrame(F, kc), ldKxN(sEnc, kc));
#pragma unroll
    for (int v = 0; v < 8; ++v) {
      const float u = x[v];
      x[v] = 0.5f * u * (1.f + erff(u * 0.70710678118654752f));
    }
    // layernorm over channels: reduce across the 16-lane half (cols of C tile)
#pragma unroll
    for (int v = 0; v < 8; ++v) {
      const float u = x[v];
      float s = u;
#pragma unroll
      for (int m = 1; m < 16; m <<= 1) s += __shfl_xor(s, m, 16);
      const float mu = s * 0.0625f;
      const float d = u - mu;
      float s2 = d * d;
#pragma unroll
      for (int m = 1; m < 16; m <<= 1) s2 += __shfl_xor(s2, m, 16);
      x[v] = d * rsqrtf(s2 * 0.0625f + 1e-5f) * lnw + lnb;
    }
    stC(SCR, x);  // normalized features [n][c]

    v8f kkc = splat8(0.f);
    MM16(kkc, ldMxK(SCR, kc), ldKxN(sWk, kc));
    stC(KK, kkc);
    v8f vvc = splat8(0.f);
    MM16(vvc, ldMxK(SCR, kc), ldKxN(sWv, kc));
    stC(VV, vvc);

#pragma unroll 1
    for (int it = 0; it < 3; ++it) {
      // q = s @ wq^T  (slot rows 0..3)
      v8f qc = splat8(0.f);
      MM16(qc, ldPad4(SL, kc), ldKxN(sWq, kc));
      stC4(SCR, qc);
      // attn = (q @ kk^T) * scale ; softmax over the 4 slots per column
      v8f at = splat8(0.f);
      MM16(at, ldPad4(SCR, kc), ldMxK(KK, kc));  // B = KK^T
      {
        float e0 = at[0] * 0.25f, e1 = at[1] * 0.25f;
        float e2 = at[2] * 0.25f, e3 = at[3] * 0.25f;
        const float mx = fmaxf(fmaxf(e0, e1), fmaxf(e2, e3));
        e0 = __expf(e0 - mx); e1 = __expf(e1 - mx);
        e2 = __expf(e2 - mx); e3 = __expf(e3 - mx);
        const float inv = 1.f / (e0 + e1 + e2 + e3);
        at[0] = e0 * inv; at[1] = e1 * inv; at[2] = e2 * inv; at[3] = e3 * inv;
      }
      stC4(SCR, at);
      // upd = attn @ vv
      v8f up = splat8(0.f);
      MM16(up, ldPad4(SCR, kc), ldKxN(VV, kc));
      stC4(SCR, up);
      // GRU cell (torch semantics), gates r,z,n
      v8f gi = splat8(bihr), gh = splat8(bhhr);
      MM16(gi, ldPad4(SCR, kc), ldKxN(sWih + 0, kc));
      MM16(gh, ldPad4(SL, kc), ldKxN(sWhh + 0, kc));
      v8f r;
#pragma unroll
      for (int v = 0; v < 8; ++v) r[v] = sigm(gi[v] + gh[v]);
      gi = splat8(bihz);
      gh = splat8(bhhz);
      MM16(gi, ldPad4(SCR, kc), ldKxN(sWih + 256, kc));
      MM16(gh, ldPad4(SL, kc), ldKxN(sWhh + 256, kc));
      v8f z;
#pragma unroll
      for (int v = 0; v < 8; ++v) z[v] = sigm(gi[v] + gh[v]);
      gi = splat8(bihn);
      gh = splat8(bhhn);
      MM16(gi, ldPad4(SCR, kc), ldKxN(sWih + 512, kc));
      MM16(gh, ldPad4(SL, kc), ldKxN(sWhh + 512, kc));
      if (lane < 16) {
#pragma unroll
        for (int v = 0; v < 4; ++v) {
          const float nn = tanhf(gi[v] + r[v] * gh[v]);
          const float h = SL[v * 16 + l16];
          SL[v * 16 + l16] = (1.f - z[v]) * nn + z[v] * h;
        }
      }
    }
    // spat = s @ s2s_w^T in 16 column chunks; scatter into mixed [64ch][16hw]
#pragma unroll 1
    for (int q = 0; q < 16; ++q) {
      v8f sp = splat8(s2s_b[q * 16 + l16]);
      MM16(sp, ldPad4(SL, kc), ldS2s(s2s_w, q, kc));
      if (lane < 16) {
#pragma unroll
        for (int v = 0; v < 4; ++v) MXD[(v * 16 + q) * 16 + l16] = sp[v];
      }
    }
    // mix: out16x16 = mix_w[16x64] @ mixed[64x16]
    v8f mc;
#pragma unroll
    for (int v = 0; v < 8; ++v) mc[v] = mixb0[v];
    MM64(mc, ldMxK64(sMix, kc), ldKxN(MXD, kc));
    stC(SCR, mc);
    // decoder (transposed conv k4 s4): pred = sigmoid(mixed^T @ dec_w + b)
    v8f pc = splat8(decb);
    MM16(pc, ldKxN(SCR, kc), ldKxN(sDec, kc));  // A = SCR^T [hw][c]
#pragma unroll
    for (int v = 0; v < 8; ++v) {
      const int m = v + mo;  // spatial i*4+j
      const int n = l16;     // d*4+e
      const int row = 4 * (m >> 2) + (n >> 2);
      const int col = 4 * (m & 3) + (n & 3);
      MXD[row * 16 + col] = sigm(pc[v]);
    }
    {  // coalesced prediction store
      float* ob = out + (b * 6 + (size_t)t) * 256;
      const float4* m4 = reinterpret_cast<const float4*>(MXD);
      float4* o4 = reinterpret_cast<float4*>(ob);
      o4[lane * 2 + 0] = m4[lane * 2 + 0];
      o4[lane * 2 + 1] = m4[lane * 2 + 1];
    }
  }
}

extern "C" void kernel_launch(void* const* d_in, const int* in_sizes, int n_in,
                              void* d_out, int out_size, void* d_ws,
                              size_t ws_size, hipStream_t stream) {
  (void)in_sizes; (void)n_in; (void)out_size; (void)d_ws; (void)ws_size;
  const float* frames = (const float*)d_in[0];
  const float* enc_w = (const float*)d_in[1];
  const float* enc_b = (const float*)d_in[2];
  const float* ln_w = (const float*)d_in[3];
  const float* ln_b = (const float*)d_in[4];
  const float* slot_mu = (const float*)d_in[5];
  const float* wq = (const float*)d_in[6];
  const float* wk = (const float*)d_in[7];
  const float* wv = (const float*)d_in[8];
  const float* gru_wih = (const float*)d_in[9];
  const float* gru_whh = (const float*)d_in[10];
  const float* gru_bih = (const float*)d_in[11];
  const float* gru_bhh = (const float*)d_in[12];
  const float* s2s_w = (const float*)d_in[13];
  const float* s2s_b = (const float*)d_in[14];
  const float* mix_w = (const float*)d_in[15];
  const float* mix_b = (const float*)d_in[16];
  const float* dec_w = (const float*)d_in[17];
  const float* dec_b = (const float*)d_in[18];

  dim3 grid(16384 / 4);
  dim3 block(128);
  microslot_kernel<<<grid, block, 0, stream>>>(
      frames, enc_w, enc_b, ln_w, ln_b, slot_mu, wq, wk, wv, gru_wih, gru_whh,
      gru_bih, gru_bhh, s2s_w, s2s_b, mix_w, mix_b, dec_w, dec_b,
      (float*)d_out);
}